// FastExtendedKalmanFilter_39178691674376
// MI455X (gfx1250) — compile-verified
//
#include <hip/hip_runtime.h>

typedef __attribute__((ext_vector_type(2))) float v2f;
typedef __attribute__((ext_vector_type(8))) float v8f;

#define TB 256
#define BS 64   // Cholesky panel / GEMM macro-tile edge

// ---------------- small prep kernels ----------------

__global__ void k_prep(const float* __restrict__ F, const float* __restrict__ Sig0,
                       const float* __restrict__ Cu, const float* __restrict__ Cw,
                       const float* __restrict__ s0,
                       float* __restrict__ s, float* __restrict__ sig,
                       float* __restrict__ scal, int E) {
  int e = blockIdx.x * blockDim.x + threadIdx.x;
  if (e < E) {
    size_t d = (size_t)e * E + e;          // diagonal element
    float f = F[d];
    s[e]   = f * s0[e];                    // s = F s0   (F diagonal)
    sig[e] = f * f * Sig0[d] + Cu[d];      // Sigma diag = F^2*Sig0 + Cu
  }
  if (blockIdx.x == 0 && threadIdx.x == 0) scal[0] = Cw[0]; // c_w
}

__global__ void k_zero(float* __restrict__ p, size_t n) {
  size_t i = (size_t)blockIdx.x * blockDim.x + threadIdx.x;
  size_t st = (size_t)gridDim.x * blockDim.x;
  for (; i < n; i += st) p[i] = 0.f;
}

// L = B diag(s) B^T built sparsely from edge list
__global__ void k_buildL(const int* __restrict__ edges, const float* __restrict__ s,
                         float* __restrict__ L, int E, int N) {
  int e = blockIdx.x * blockDim.x + threadIdx.x;
  if (e >= E) return;
  int n = edges[2 * e], m = edges[2 * e + 1];
  float se = s[e];
  atomicAdd(&L[(size_t)n * N + n],  se);
  atomicAdd(&L[(size_t)m * N + m],  se);
  atomicAdd(&L[(size_t)n * N + m], -se);
  atomicAdd(&L[(size_t)m * N + n], -se);
}

// y = L x  (one block per row)
__global__ void k_matvec(const float* __restrict__ L, const float* __restrict__ x,
                         float* __restrict__ y, int N) {
  __shared__ float red[TB];
  int row = blockIdx.x;
  float acc = 0.f;
  for (int j = threadIdx.x; j < N; j += TB) acc += L[(size_t)row * N + j] * x[j];
  red[threadIdx.x] = acc;
  __syncthreads();
  for (int st = TB / 2; st > 0; st >>= 1) {
    if (threadIdx.x < st) red[threadIdx.x] += red[threadIdx.x + st];
    __syncthreads();
  }
  if (threadIdx.x == 0) y[row] = red[0];
}

// dc[p,e] = c[p, n_e] - c[p, m_e]
__global__ void k_dc(const float* __restrict__ c, const int* __restrict__ edges,
                     float* __restrict__ dc, int E, int N, int P) {
  int e = blockIdx.x * blockDim.x + threadIdx.x;
  if (e >= E) return;
  int n = edges[2 * e], m = edges[2 * e + 1];
  for (int p = 0; p < P; ++p)
    dc[(size_t)p * E + e] = c[(size_t)p * N + n] - c[(size_t)p * N + m];
}

// r = y - sum_p a_p c[p]
__global__ void k_resid(const float* __restrict__ c, const float* __restrict__ a,
                        const float* __restrict__ y, float* __restrict__ r,
                        int N, int P) {
  int n = blockIdx.x * blockDim.x + threadIdx.x;
  if (n >= N) return;
  float f = 0.f;
  for (int p = 0; p < P; ++p) f += a[p] * c[(size_t)p * N + n];
  r[n] = y[n] - f;
}

// G2 = a[P-1]*L + a[P-2]*I
__global__ void k_g2(const float* __restrict__ L, const float* __restrict__ a,
                     float* __restrict__ G2, int N, int P) {
  size_t i = (size_t)blockIdx.x * blockDim.x + threadIdx.x;
  size_t nn = (size_t)N * N;
  if (i >= nn) return;
  int row = (int)(i / N), col = (int)(i % N);
  G2[i] = a[P - 1] * L[i] + ((row == col) ? a[P - 2] : 0.f);
}

// ---------------- register-blocked WMMA fp32 GEMM ----------------
// Each wave computes a 64x64 macro-tile as 4x4 WMMA 16x16 subtiles (K step 4).
//   TRANSB=0: Bop[k,n] = B[k*ldb+n]
//   TRANSB=1: Bop[k,n] = B[n*ldb+k]   (optionally * scaleK[k] when SCALEB)
//   SUB=0:    C = A@Bop  (+ diagAdd[0] on global diagonal rr==cc)
//   SUB=1:    C = C - A@Bop           (Cholesky SYRK trailing update)
template <bool TRANSB, bool SCALEB, bool SUB>
__global__ __launch_bounds__(256) void k_gemm_wmma(
    const float* __restrict__ A, int lda, const float* __restrict__ B, int ldb,
    const float* __restrict__ scaleK, const float* __restrict__ diagAdd,
    float* __restrict__ C, int ldc, int M, int Nc, int K) {
  int lane = threadIdx.x & 31;
  int wid  = threadIdx.x >> 5;
  int tN = Nc >> 6;
  int tM = M >> 6;
  int tile = blockIdx.x * 8 + wid;
  if (tile >= tM * tN) return;               // wave-uniform: EXEC stays all-ones
  int row0 = (tile / tN) << 6;
  int col0 = (tile % tN) << 6;
  int h = lane >> 4;                          // lane half -> K offset {2h, 2h+1}
  int l = lane & 15;

  v8f acc[4][4];
#pragma unroll
  for (int i = 0; i < 4; ++i)
#pragma unroll
    for (int j = 0; j < 4; ++j) acc[i][j] = (v8f){0.f,0.f,0.f,0.f,0.f,0.f,0.f,0.f};

  const float* Ab[4];
#pragma unroll
  for (int i = 0; i < 4; ++i)
    Ab[i] = A + (size_t)(row0 + 16 * i + l) * lda + 2 * h;
  const float* Bb[4];
  if (TRANSB) {
#pragma unroll
    for (int j = 0; j < 4; ++j)
      Bb[j] = B + (size_t)(col0 + 16 * j + l) * ldb + 2 * h;
  }

  for (int k = 0; k < K; k += 4) {
    v2f av[4];
#pragma unroll
    for (int i = 0; i < 4; ++i) av[i] = *(const v2f*)(Ab[i] + k);
    v2f bv[4];
    if (TRANSB) {
      v2f sc;
      if (SCALEB) sc = *(const v2f*)(scaleK + k + 2 * h);
#pragma unroll
      for (int j = 0; j < 4; ++j) {
        bv[j] = *(const v2f*)(Bb[j] + k);
        if (SCALEB) bv[j] *= sc;
      }
    } else {
#pragma unroll
      for (int j = 0; j < 4; ++j) {
        bv[j].x = B[(size_t)(k + 2 * h)     * ldb + col0 + 16 * j + l];
        bv[j].y = B[(size_t)(k + 2 * h + 1) * ldb + col0 + 16 * j + l];
      }
    }
#pragma unroll
    for (int i = 0; i < 4; ++i)
#pragma unroll
      for (int j = 0; j < 4; ++j)
        acc[i][j] = __builtin_amdgcn_wmma_f32_16x16x4_f32(
            false, av[i], false, bv[j], (short)0, acc[i][j], false, false);
  }

  float dv = (!SUB && diagAdd) ? diagAdd[0] : 0.f;
#pragma unroll
  for (int i = 0; i < 4; ++i)
#pragma unroll
    for (int j = 0; j < 4; ++j)
#pragma unroll
      for (int v = 0; v < 8; ++v) {
        int rr = row0 + 16 * i + v + 8 * h;   // C/D layout: vgpr v -> M=v (+8 hi half)
        int cc = col0 + 16 * j + l;
        size_t idx = (size_t)rr * ldc + cc;
        if (SUB) C[idx] -= acc[i][j][v];
        else     C[idx] = acc[i][j][v] + ((rr == cc) ? dv : 0.f);
      }
}

// H[n,e] = dc0*(G4 col-diff) + dc1*(G3 col-diff) + dc2*(G2 col-diff) + a[P-1]*dc[P-2]*(delta diff)
__global__ void k_buildH(const float* __restrict__ G2, const float* __restrict__ G3,
                         const float* __restrict__ G4, const float* __restrict__ dc,
                         const int* __restrict__ edges, const float* __restrict__ a,
                         float* __restrict__ H, int N, int E, int P) {
  size_t i = (size_t)blockIdx.x * blockDim.x + threadIdx.x;
  if (i >= (size_t)N * E) return;
  int n = (int)(i / E), e = (int)(i % E);
  int ne = edges[2 * e], me = edges[2 * e + 1];
  size_t rb = (size_t)n * N;
  float hv = dc[(size_t)0 * E + e] * (G4[rb + ne] - G4[rb + me])
           + dc[(size_t)1 * E + e] * (G3[rb + ne] - G3[rb + me])
           + dc[(size_t)2 * E + e] * (G2[rb + ne] - G2[rb + me]);
  float aP = a[P - 1];
  hv += aP * dc[(size_t)(P - 2) * E + e] *
        ((n == ne ? 1.f : 0.f) - (n == me ? 1.f : 0.f));
  H[i] = hv;
}

// Blocked Cholesky: single-block panel factorization of columns [k0, k0+BS)
// over rows [k0, N). Trailing update is done by the WMMA SYRK between panels.
__global__ __launch_bounds__(1024) void k_chol_panel(float* __restrict__ S,
                                                     int N, int k0) {
  const int t = threadIdx.x, nt = blockDim.x;
  for (int j = 0; j < BS; ++j) {
    int c = k0 + j;
    if (t == 0) S[(size_t)c * N + c] = sqrtf(S[(size_t)c * N + c]);
    __syncthreads();
    float d = S[(size_t)c * N + c];
    for (int i = c + 1 + t; i < N; i += nt) S[(size_t)i * N + c] /= d;
    __syncthreads();
    int rem  = BS - 1 - j;           // panel columns still to update
    int rows = N - (c + 1);
    int total = rem * rows;
    for (int w = t; w < total; w += nt) {
      int jj = w / rows;
      int i  = c + 1 + (w - jj * rows);
      int cc = c + 1 + jj;
      S[(size_t)i * N + cc] -= S[(size_t)i * N + c] * S[(size_t)cc * N + c];
    }
    __syncthreads();
  }
}

// forward/back substitution with the Cholesky factor (lower triangle of S): z = S^{-1} r
__global__ __launch_bounds__(1024) void k_trisolve(const float* __restrict__ S,
                                                   float* __restrict__ r,
                                                   float* __restrict__ z, int N) {
  const int t = threadIdx.x, nt = blockDim.x;
  for (int k = 0; k < N; ++k) {             // L w = r
    if (t == 0) r[k] /= S[(size_t)k * N + k];
    __syncthreads();
    float wk = r[k];
    for (int i = k + 1 + t; i < N; i += nt) r[i] -= S[(size_t)i * N + k] * wk;
    __syncthreads();
  }
  for (int k = N - 1; k >= 0; --k) {        // L^T z = w
    if (t == 0) r[k] /= S[(size_t)k * N + k];
    __syncthreads();
    float xk = r[k];
    for (int i = t; i < k; i += nt) r[i] -= S[(size_t)k * N + i] * xk;
    __syncthreads();
  }
  for (int i = t; i < N; i += nt) z[i] = r[i];
}

// s_out[e] = relu( s[e] + sig[e] * (H^T z)[e] )
__global__ void k_out(const float* __restrict__ H, const float* __restrict__ z,
                      const float* __restrict__ s, const float* __restrict__ sig,
                      float* __restrict__ out, int N, int E) {
  int e = blockIdx.x * blockDim.x + threadIdx.x;
  if (e >= E) return;
  float acc = 0.f;
  for (int n = 0; n < N; ++n) acc += H[(size_t)n * E + e] * z[n];
  float v = s[e] + sig[e] * acc;
  out[e] = v > 0.f ? v : 0.f;
}

// ---------------- host orchestration ----------------

extern "C" void kernel_launch(void* const* d_in, const int* in_sizes, int n_in,
                              void* d_out, int out_size, void* d_ws, size_t ws_size,
                              hipStream_t stream) {
  const float* q    = (const float*)d_in[0];
  const float* y    = (const float*)d_in[1];
  const float* F    = (const float*)d_in[2];
  const int*   edges= (const int*)  d_in[4];
  const float* Cu   = (const float*)d_in[5];
  const float* Cw   = (const float*)d_in[6];
  const float* s0   = (const float*)d_in[7];
  const float* Sig0 = (const float*)d_in[8];
  const float* a    = (const float*)d_in[9];
  const int N = in_sizes[0];   // 1024
  const int E = in_sizes[7];   // 4096
  const int P = in_sizes[9];   // 5

  float* ws = (float*)d_ws;
  size_t off = 0;
  float* s    = ws + off; off += E;
  float* sig  = ws + off; off += E;
  float* scal = ws + off; off += 16;
  float* c    = ws + off; off += (size_t)P * N;
  float* dc   = ws + off; off += (size_t)P * E;
  float* r    = ws + off; off += N;
  float* z    = ws + off; off += N;
  off = (off + 63) & ~(size_t)63;
  float* L    = ws + off; off += (size_t)N * N;
  float* G2   = ws + off; off += (size_t)N * N;
  float* G3   = ws + off; off += (size_t)N * N;
  float* G4   = ws + off; off += (size_t)N * N;
  float* H    = ws + off; off += (size_t)N * E;
  float* Smat = ws + off; off += (size_t)N * N;
  (void)ws_size; (void)n_in; (void)out_size;

  dim3 b(TB);
  k_prep<<<dim3((E + TB - 1) / TB), b, 0, stream>>>(F, Sig0, Cu, Cw, s0, s, sig, scal, E);
  k_zero<<<dim3(256), b, 0, stream>>>(L, (size_t)N * N);
  k_buildL<<<dim3((E + TB - 1) / TB), b, 0, stream>>>(edges, s, L, E, N);

  hipMemcpyAsync(c, q, (size_t)N * sizeof(float), hipMemcpyDeviceToDevice, stream);
  for (int p = 1; p < P; ++p)
    k_matvec<<<dim3(N), b, 0, stream>>>(L, c + (size_t)(p - 1) * N, c + (size_t)p * N, N);

  k_dc<<<dim3((E + TB - 1) / TB), b, 0, stream>>>(c, edges, dc, E, N, P);
  k_resid<<<dim3((N + TB - 1) / TB), b, 0, stream>>>(c, a, y, r, N, P);
  k_g2<<<dim3((int)(((size_t)N * N + TB - 1) / TB)), b, 0, stream>>>(L, a, G2, N, P);

  int tilesNN = (N / BS) * (N / BS);
  int gblk = (tilesNN + 7) / 8;
  // G3 = L @ G2 + a[P-3] I ;  G4 = L @ G3 + a[P-4] I
  k_gemm_wmma<false,false,false><<<dim3(gblk), b, 0, stream>>>(
      L, N, G2, N, nullptr, a + (P - 3), G3, N, N, N, N);
  k_gemm_wmma<false,false,false><<<dim3(gblk), b, 0, stream>>>(
      L, N, G3, N, nullptr, a + (P - 4), G4, N, N, N, N);

  k_buildH<<<dim3((int)(((size_t)N * E + TB - 1) / TB)), b, 0, stream>>>(
      G2, G3, G4, dc, edges, a, H, N, E, P);

  // S = H diag(sig) H^T + c_w I
  k_gemm_wmma<true,true,false><<<dim3(gblk), b, 0, stream>>>(
      H, E, H, E, sig, scal, Smat, N, N, N, E);

  // Blocked Cholesky: panel factor (1 block) + WMMA SYRK trailing update
  for (int k0 = 0; k0 < N; k0 += BS) {
    k_chol_panel<<<dim3(1), dim3(1024), 0, stream>>>(Smat, N, k0);
    int Mt = N - k0 - BS;
    if (Mt > 0) {
      int tiles = (Mt / BS) * (Mt / BS);
      int blk = (tiles + 7) / 8;
      const float* L21 = Smat + (size_t)(k0 + BS) * N + k0;
      float* S22 = Smat + (size_t)(k0 + BS) * N + (k0 + BS);
      k_gemm_wmma<true,false,true><<<dim3(blk), b, 0, stream>>>(
          L21, N, L21, N, nullptr, nullptr, S22, N, Mt, Mt, BS);
    }
  }
  k_trisolve<<<dim3(1), dim3(1024), 0, stream>>>(Smat, r, z, N);
  k_out<<<dim3((E + TB - 1) / TB), b, 0, stream>>>(H, z, s, sig, (float*)d_out, N, E);
}